// AttentionHead_6253472383029
// MI455X (gfx1250) — compile-verified
//
#include <hip/hip_runtime.h>
#include <stdint.h>

// ---------------------------------------------------------------------------
// AttentionHead for MI455X (gfx1250), fp32 end-to-end via V_WMMA_F32_16X16X4_F32
//   B=8, S=2048, DK=512
//   qh = q@Wq^T+bq ; kh,vh likewise        (projection GEMMs, WMMA f32)
//   scores = qh @ (kh memory as [512x2048]) * sqrt(512)
//   out    = softmax(scores) @ vh          (flash-style streaming softmax)
// CDNA5 paths: v_wmma_f32_16x16x4_f32 matrix core, double-buffered
// global_load_async_to_lds_b128 staging (ASYNCcnt), global_prefetch_b8.
// ---------------------------------------------------------------------------

typedef __attribute__((ext_vector_type(2))) float v2f;
typedef __attribute__((ext_vector_type(8))) float v8f;

#define WMMA_F32(a, b, c) \
  __builtin_amdgcn_wmma_f32_16x16x4_f32(false, (a), false, (b), (short)0, (c), false, false)

// Async 16B copy global -> LDS, tracked by ASYNCcnt (CDNA5 §10 async ops).
#define ASYNC_COPY_B128(ldsOff, gaddr)                                   \
  asm volatile("global_load_async_to_lds_b128 %0, %1, off" ::            \
                   "v"(ldsOff), "v"(gaddr)                               \
               : "memory")
#define ASYNC_WAIT_8() asm volatile("s_wait_asynccnt 8" ::: "memory")
#define ASYNC_WAIT_0() asm volatile("s_wait_asynccnt 0" ::: "memory")

static __device__ __forceinline__ uint32_t lds_off(const void* p) {
  return (uint32_t)(uintptr_t)p;  // flat LDS aperture: addr[31:0] = LDS byte offset
}

#define B_DIM 8
#define S_DIM 2048
#define DK_DIM 512

// ---------------------------------------------------------------------------
// Projection GEMM: out[m,n] = sum_k x[m,k] * W[n,k] + bias[n]
// M x 512 x 512, block tile 64x64, 4 waves, K staged 16-wide through LDS.
// ---------------------------------------------------------------------------
__global__ __launch_bounds__(128)
void proj_kernel(const float* __restrict__ x,     // [M, 512]
                 const float* __restrict__ W,     // [512, 512] (out x in)
                 const float* __restrict__ bias,  // [512]
                 float* __restrict__ out)         // [M, 512]
{
  __shared__ float As[64 * 17];
  __shared__ float Bs[16 * 68];

  const int tid  = threadIdx.x;
  const int wave = tid >> 5;
  const int lane = tid & 31;
  const int half = lane >> 4;
  const int l16  = lane & 15;
  const int m0   = blockIdx.y * 64;
  const int n0   = blockIdx.x * 64;

  v8f acc[4] = {};

  for (int k0 = 0; k0 < DK_DIM; k0 += 16) {
    for (int t = tid; t < 256; t += 128) {
      const int c = t & 3, m = t >> 2;
      const float4 va = *(const float4*)(x + (size_t)(m0 + m) * DK_DIM + k0 + c * 4);
      float* dst = &As[m * 17 + c * 4];
      dst[0] = va.x; dst[1] = va.y; dst[2] = va.z; dst[3] = va.w;
    }
    for (int t = tid; t < 256; t += 128) {
      const int c = t & 3, n = t >> 2;
      const float4 vb = *(const float4*)(W + (size_t)(n0 + n) * DK_DIM + k0 + c * 4);
      Bs[(c * 4 + 0) * 68 + n] = vb.x;
      Bs[(c * 4 + 1) * 68 + n] = vb.y;
      Bs[(c * 4 + 2) * 68 + n] = vb.z;
      Bs[(c * 4 + 3) * 68 + n] = vb.w;
    }
    __syncthreads();

#pragma unroll
    for (int kk = 0; kk < 16; kk += 4) {
      v2f a;
      a.x = As[(wave * 16 + l16) * 17 + kk + 2 * half];
      a.y = As[(wave * 16 + l16) * 17 + kk + 2 * half + 1];
#pragma unroll
      for (int nt = 0; nt < 4; ++nt) {
        v2f b;
        b.x = Bs[(kk + 2 * half) * 68 + nt * 16 + l16];
        b.y = Bs[(kk + 2 * half + 1) * 68 + nt * 16 + l16];
        acc[nt] = WMMA_F32(a, b, acc[nt]);
      }
    }
    __syncthreads();
  }

#pragma unroll
  for (int nt = 0; nt < 4; ++nt) {
    const int col = n0 + nt * 16 + l16;
    const float bv = bias[col];
#pragma unroll
    for (int r = 0; r < 8; ++r) {
      const int row = m0 + wave * 16 + r + 8 * half;
      out[(size_t)row * DK_DIM + col] = acc[nt][r] + bv;
    }
  }
}

// ---------------------------------------------------------------------------
// Flash attention: block = (batch, 32 query rows), 8 waves (256 threads).
// kh/vh tiles double-buffered through LDS via global_load_async_to_lds_b128.
// ---------------------------------------------------------------------------
#define QS_STRIDE 516   // 512+4 pad -> bank-conflict-free column reads
#define PS_STRIDE 132   // 128+4 pad
#define KS_STRIDE 132   // kh chunk: 64 rows x 128 cols (rows 2 apart -> +8 banks)
#define VS_STRIDE 520   // vh chunk: 16 rows x 512 cols (rows 2 apart -> +16 banks)
#define STAGE_HALF 8448 // floats per staging buffer (= 64*132; >= 16*520)

__global__ __launch_bounds__(256)
void attn_kernel(const float* __restrict__ qh,   // [B, S, 512]
                 const float* __restrict__ kh,   // per-batch viewed as [512][2048]
                 const float* __restrict__ vh,   // [B, S, 512]
                 float* __restrict__ out)        // [B, S, 512]
{
  extern __shared__ float smem[];
  float* Qs       = smem;                     // 32 x QS_STRIDE   (16512)
  float* Ps       = Qs + 32 * QS_STRIDE;      // 32 x PS_STRIDE   ( 4224)
  float* rowmax   = Ps + 32 * PS_STRIDE;      // 32
  float* rowsum   = rowmax + 32;              // 32
  float* rowscale = rowsum + 32;              // 32
  float* red      = rowscale + 32;            // 256
  float* stage0   = red + 256;                // STAGE_HALF
  float* stage1   = stage0 + STAGE_HALF;      // STAGE_HALF

  const int tid  = threadIdx.x;
  const int wave = tid >> 5;
  const int lane = tid & 31;
  const int half = lane >> 4;
  const int l16  = lane & 15;
  const int bb   = blockIdx.y;
  const int q0   = blockIdx.x * 32;

  const float* qb  = qh + (size_t)bb * S_DIM * DK_DIM;
  const float* khb = kh + (size_t)bb * S_DIM * DK_DIM;  // KR [512][2048]
  const float* vhb = vh + (size_t)bb * S_DIM * DK_DIM;  // [2048][512]

  // stage the 32x512 qh tile (plain loads; one-time)
  for (int t = tid; t < 32 * 128; t += 256) {
    const int row = t >> 7;
    const int c4  = t & 127;
    const float4 v = *(const float4*)(qb + (size_t)(q0 + row) * DK_DIM + c4 * 4);
    *(float4*)(&Qs[row * QS_STRIDE + c4 * 4]) = v;
  }
  if (tid < 32) {
    rowmax[tid]   = -__builtin_inff();
    rowsum[tid]   = 0.0f;
    rowscale[tid] = 0.0f;
  }
  __syncthreads();

  const int rt = wave >> 2;  // 16-row subtile (0/1)
  const int ct = wave & 3;   // score 32-col subtile
  const int cb = wave & 3;   // output 128-col block

  v8f oacc[8] = {};
  const float SCALE = 22.62741699796952f;  // sqrt(512) (kept source "bug")

  const float* qrow  = &Qs[(rt * 16 + l16) * QS_STRIDE];
  const float* prow2 = &Ps[(rt * 16 + l16) * PS_STRIDE];

  for (int j = 0; j < S_DIM; j += 128) {
    // warm the vh tile for this iteration's output phase
    __builtin_prefetch(vhb + (size_t)j * DK_DIM + tid * 256, 0, 1);

    // ============ score phase: S16x128 = Q(16x512) @ KR(512x128) ============
    v8f sacc0 = {}, sacc1 = {};
    // preload k-chunk 0 (KR[0:64][j:j+128])
    for (int t = tid; t < 2048; t += 256) {
      const int r = t >> 5, c4 = (t & 31) * 4;
      ASYNC_COPY_B128(lds_off(stage0 + r * KS_STRIDE + c4),
                      (uint64_t)(uintptr_t)(khb + (size_t)r * S_DIM + j + c4));
    }
    for (int c = 0; c < 8; ++c) {
      float* buf = (c & 1) ? stage1 : stage0;
      if (c + 1 < 8) {  // issue chunk c+1 into the other buffer
        float* nbuf = (c & 1) ? stage0 : stage1;
        const int k0n = (c + 1) * 64;
        for (int t = tid; t < 2048; t += 256) {
          const int r = t >> 5, c4 = (t & 31) * 4;
          ASYNC_COPY_B128(lds_off(nbuf + r * KS_STRIDE + c4),
                          (uint64_t)(uintptr_t)(khb + (size_t)(k0n + r) * S_DIM + j + c4));
        }
        ASYNC_WAIT_8();  // chunk c landed (in-order); chunk c+1 may be in flight
      } else {
        ASYNC_WAIT_0();
      }
      __syncthreads();  // all waves' copies for chunk c visible

      const int k0 = c * 64;
#pragma unroll 4
      for (int kk = 0; kk < 64; kk += 4) {
        const int krow = kk + 2 * half;
        v2f a, b0, b1;
        a.x  = qrow[k0 + krow];
        a.y  = qrow[k0 + krow + 1];
        b0.x = buf[krow * KS_STRIDE + ct * 32 + l16];
        b0.y = buf[(krow + 1) * KS_STRIDE + ct * 32 + l16];
        b1.x = buf[krow * KS_STRIDE + ct * 32 + 16 + l16];
        b1.y = buf[(krow + 1) * KS_STRIDE + ct * 32 + 16 + l16];
        sacc0 = WMMA_F32(a, b0, sacc0);
        sacc1 = WMMA_F32(a, b1, sacc1);
      }
      __syncthreads();  // done reading buf before it is restaged
    }
#pragma unroll
    for (int r = 0; r < 8; ++r) {
      const int prow = rt * 16 + r + 8 * half;
      Ps[prow * PS_STRIDE + ct * 32 + l16]      = sacc0[r] * SCALE;
      Ps[prow * PS_STRIDE + ct * 32 + 16 + l16] = sacc1[r] * SCALE;
    }
    __syncthreads();

    // ============ streaming softmax over the 32x128 tile ============
    {
      const int row = tid >> 3;
      const int seg = tid & 7;
      float* pr = &Ps[row * PS_STRIDE + seg * 16];
      float lmax = -__builtin_inff();
#pragma unroll
      for (int c = 0; c < 16; ++c) lmax = fmaxf(lmax, pr[c]);
      red[row * 8 + seg] = lmax;
      __syncthreads();
      if (seg == 0) {
        const float mold = rowmax[row];
        float tm = red[row * 8];
#pragma unroll
        for (int s2 = 1; s2 < 8; ++s2) tm = fmaxf(tm, red[row * 8 + s2]);
        const float mnew = fmaxf(mold, tm);
        rowscale[row] = __expf(mold - mnew);  // first iter: exp(-inf)=0
        rowmax[row]   = mnew;
      }
      __syncthreads();
      const float mnew = rowmax[row];
      float psum = 0.0f;
#pragma unroll
      for (int c = 0; c < 16; ++c) {
        const float e = __expf(pr[c] - mnew);
        pr[c] = e;
        psum += e;
      }
      red[row * 8 + seg] = psum;
      __syncthreads();
      if (seg == 0) {
        float s2 = 0.0f;
#pragma unroll
        for (int qq = 0; qq < 8; ++qq) s2 += red[row * 8 + qq];
        rowsum[row] = rowsum[row] * rowscale[row] + s2;
      }
      __syncthreads();
    }

    // ============ output phase: rescale + P(16x128) @ V(128x128) ============
#pragma unroll
    for (int nt = 0; nt < 8; ++nt) {
#pragma unroll
      for (int r = 0; r < 8; ++r) {
        oacc[nt][r] *= rowscale[rt * 16 + r + 8 * half];
      }
    }
    // preload v-chunk 0 (vh[j:j+16][0:512])
    for (int t = tid; t < 2048; t += 256) {
      const int r = t >> 7, c4 = (t & 127) * 4;
      ASYNC_COPY_B128(lds_off(stage0 + r * VS_STRIDE + c4),
                      (uint64_t)(uintptr_t)(vhb + (size_t)(j + r) * DK_DIM + c4));
    }
    for (int c = 0; c < 8; ++c) {
      float* buf = (c & 1) ? stage1 : stage0;
      if (c + 1 < 8) {
        float* nbuf = (c & 1) ? stage0 : stage1;
        const int v0n = (c + 1) * 16;
        for (int t = tid; t < 2048; t += 256) {
          const int r = t >> 7, c4 = (t & 127) * 4;
          ASYNC_COPY_B128(lds_off(nbuf + r * VS_STRIDE + c4),
                          (uint64_t)(uintptr_t)(vhb + (size_t)(j + v0n + r) * DK_DIM + c4));
        }
        ASYNC_WAIT_8();
      } else {
        ASYNC_WAIT_0();
      }
      __syncthreads();

      const int v0 = c * 16;
#pragma unroll
      for (int kk = 0; kk < 16; kk += 4) {
        const int krow = kk + 2 * half;
        v2f a;
        a.x = prow2[v0 + krow];
        a.y = prow2[v0 + krow + 1];
#pragma unroll
        for (int nt = 0; nt < 8; ++nt) {
          v2f bvv;
          bvv.x = buf[krow * VS_STRIDE + cb * 128 + nt * 16 + l16];
          bvv.y = buf[(krow + 1) * VS_STRIDE + cb * 128 + nt * 16 + l16];
          oacc[nt] = WMMA_F32(a, bvv, oacc[nt]);
        }
      }
      __syncthreads();
    }
  }

  // ============ finalize: divide by softmax denominator, store ============
#pragma unroll
  for (int nt = 0; nt < 8; ++nt) {
#pragma unroll
    for (int r = 0; r < 8; ++r) {
      const int lrow  = rt * 16 + r + 8 * half;
      const float den = rowsum[lrow];
      const size_t orow = (size_t)bb * S_DIM + q0 + lrow;
      out[orow * DK_DIM + cb * 128 + nt * 16 + l16] = oacc[nt][r] / den;
    }
  }
}

// ---------------------------------------------------------------------------
extern "C" void kernel_launch(void* const* d_in, const int* in_sizes, int n_in,
                              void* d_out, int out_size, void* d_ws, size_t ws_size,
                              hipStream_t stream) {
  const float* q  = (const float*)d_in[0];
  const float* k  = (const float*)d_in[1];
  const float* v  = (const float*)d_in[2];
  const float* Wq = (const float*)d_in[3];
  const float* bq = (const float*)d_in[4];
  const float* Wk = (const float*)d_in[5];
  const float* bk = (const float*)d_in[6];
  const float* Wv = (const float*)d_in[7];
  const float* bv = (const float*)d_in[8];
  float* out = (float*)d_out;

  const int M = B_DIM * S_DIM;  // 16384
  float* qh = (float*)d_ws;
  float* kh = qh + (size_t)M * DK_DIM;
  float* vh = kh + (size_t)M * DK_DIM;

  dim3 pgrid(DK_DIM / 64, M / 64);
  dim3 pblock(128);
  proj_kernel<<<pgrid, pblock, 0, stream>>>(q, Wq, bq, qh);
  proj_kernel<<<pgrid, pblock, 0, stream>>>(k, Wk, bk, kh);
  proj_kernel<<<pgrid, pblock, 0, stream>>>(v, Wv, bv, vh);

  dim3 agrid(S_DIM / 32, B_DIM);
  dim3 ablock(256);
  const size_t shbytes =
      (size_t)(32 * QS_STRIDE + 32 * PS_STRIDE + 3 * 32 + 32 * 8 + 2 * STAGE_HALF) *
      sizeof(float);  // ~148 KB dynamic LDS (CDNA5 WGP has 320 KB)
  attn_kernel<<<agrid, ablock, shbytes, stream>>>(qh, kh, vh, out);
}